// GAT_Baseline_24481313587810
// MI455X (gfx1250) — compile-verified
//
#include <hip/hip_runtime.h>
#include <hip/hip_bf16.h>
#include <cstdint>

typedef __attribute__((ext_vector_type(2))) float v2f;
typedef __attribute__((ext_vector_type(8))) float v8f;

#define NEG_SLOPE 0.2f
#define BN_EPS    1e-5f
#define KDIM      256   // in_dim
#define HID       128   // hidden = heads * d_head
#define HEADS     4
#define DH        32
#define OUTD      2
#define LDS_STRIDE 260  // 256 + 4 pad: 16B-aligned rows (1040B) and 260%64=4 -> conflict-free

// ---------------------------------------------------------------------------
// float atomic max via int max / uint min trick (init must be large negative)
// ---------------------------------------------------------------------------
__device__ inline void atomicMaxF(float* addr, float val) {
  if (val >= 0.0f) {
    atomicMax((int*)addr, __float_as_int(val));
  } else {
    atomicMin((unsigned int*)addr, (unsigned int)__float_as_int(val));
  }
}

__global__ void fill_kernel(float* __restrict__ p, long long n, float v) {
  long long i = (long long)blockIdx.x * blockDim.x + threadIdx.x;
  if (i < n) p[i] = v;
}

// ---------------------------------------------------------------------------
// Layer-1 GEMM: h[N,128] = x[N,256] @ W1[256,128], fp32 WMMA 16x16x4.
// Block = 256 threads = 8 waves; block owns 16 rows, wave w owns cols [16w,16w+16).
// x tile staged to LDS via CDNA5 async DMA (GLOBAL_LOAD_ASYNC_TO_LDS_B128).
// ---------------------------------------------------------------------------
__global__ __launch_bounds__(256) void gemm1_wmma_kernel(
    const float* __restrict__ x, const float* __restrict__ W1,
    float* __restrict__ h, int N) {
  __shared__ float As[16][LDS_STRIDE];

  const int rowBase = blockIdx.x * 16;
  const int tid     = threadIdx.x;

  // Async-stage 16x256 tile of x into LDS: 1024 x 16B chunks, 4 per thread.
  // Both global (row start 1024B-aligned + 16B col) and LDS (1040B row stride
  // + 16B col) addresses are 16B aligned.
  #pragma unroll
  for (int i = 0; i < 4; ++i) {
    int chunk = i * 256 + tid;          // 0..1023
    int row   = chunk >> 6;             // 64 chunks of 4 floats per row
    int col   = (chunk & 63) << 2;      // float column, multiple of 4
    int grow  = rowBase + row;
    if (grow >= N) grow = N - 1;        // clamp: tail tiles load dup rows (unstored)
    const float* gp = x + (size_t)grow * KDIM + col;
    unsigned ldsOff = (unsigned)(uintptr_t)&As[row][col];  // low 32b of flat LDS ptr = LDS offset
    asm volatile("global_load_async_to_lds_b128 %0, %1, off"
                 :: "v"(ldsOff), "v"(gp) : "memory");
  }
  asm volatile("s_wait_asynccnt 0x0" ::: "memory");
  __syncthreads();

  const int wave    = tid >> 5;
  const int lane    = tid & 31;
  const int l15     = lane & 15;        // A: row index   B/D: col index
  const int half    = lane >> 4;        // 0: K{0,1} / M0..7   1: K{2,3} / M8..15
  const int colBase = wave * 16;

  v8f c = {};
  // K loop: 256 / 4 = 64 WMMA ops per wave
  for (int k = 0; k < KDIM; k += 4) {
    const int ka = k + half * 2;
    v2f a, b;
    // A fragment (16x4 f32): lane l15 = row M, VGPR0/1 = K={ka, ka+1}
    a[0] = As[l15][ka];
    a[1] = As[l15][ka + 1];
    // B fragment (4x16 f32): lane l15 = col N, VGPR0/1 = K={ka, ka+1}
    b[0] = W1[(size_t)ka       * HID + colBase + l15];
    b[1] = W1[(size_t)(ka + 1) * HID + colBase + l15];
    c = __builtin_amdgcn_wmma_f32_16x16x4_f32(
        /*neg_a=*/false, a, /*neg_b=*/false, b,
        /*c_mod=*/(short)0, c, /*reuse_a=*/false, /*reuse_b=*/false);
  }

  // D layout: VGPR r holds row M = r + 8*half, col = l15
  if (rowBase + 16 <= N) {              // full tile: unguarded stores
    #pragma unroll
    for (int r = 0; r < 8; ++r)
      h[(size_t)(rowBase + r + 8 * half) * HID + colBase + l15] = c[r];
  } else {
    #pragma unroll
    for (int r = 0; r < 8; ++r) {
      int gm = rowBase + r + 8 * half;
      if (gm < N) h[(size_t)gm * HID + colBase + l15] = c[r];
    }
  }
}

// ---------------------------------------------------------------------------
// Per-node attention coefficients: alpha_src/dst[n,h] = sum_d h[n,h,d]*a[h,d]
// ---------------------------------------------------------------------------
__global__ void alpha1_kernel(const float* __restrict__ h,
                              const float* __restrict__ aS,
                              const float* __restrict__ aD,
                              float* __restrict__ asrc, float* __restrict__ adst,
                              int N) {
  int i = blockIdx.x * blockDim.x + threadIdx.x;   // over N*HEADS
  if (i >= N * HEADS) return;
  int n = i >> 2, hh = i & 3;
  const float* hp = h + (size_t)n * HID + hh * DH;
  float s1 = 0.f, s2 = 0.f;
  #pragma unroll
  for (int d = 0; d < DH; ++d) {
    float v = hp[d];
    s1 = fmaf(v, aS[hh * DH + d], s1);
    s2 = fmaf(v, aD[hh * DH + d], s2);
  }
  asrc[i] = s1;
  adst[i] = s2;
}

__device__ inline void edge_nodes(const long long* __restrict__ ei,
                                  int E, long long e, int& s, int& d) {
  if (e < (long long)E) { s = (int)ei[e]; d = (int)ei[(long long)E + e]; }
  else                  { s = d = (int)(e - E); }          // self loop
}

// ---------------------------------------------------------------------------
// Edge pass A (layer 1): segment max of leaky-relu logits, per (edge, head)
// ---------------------------------------------------------------------------
__global__ void edge_max1_kernel(const long long* __restrict__ ei, int E, int N,
                                 const float* __restrict__ asrc,
                                 const float* __restrict__ adst,
                                 float* __restrict__ m1) {
  long long ET = (long long)E + N;
  long long i = (long long)blockIdx.x * blockDim.x + threadIdx.x;
  if (i >= ET * HEADS) return;
  long long e = i >> 2;
  int hh = (int)(i & 3);
  int s, d;
  edge_nodes(ei, E, e, s, d);
  float l = asrc[s * HEADS + hh] + adst[d * HEADS + hh];
  l = (l > 0.f) ? l : NEG_SLOPE * l;
  atomicMaxF(&m1[d * HEADS + hh], l);
}

// ---------------------------------------------------------------------------
// Edge pass B (layer 1): e=exp(l-m); s1[dst,h]+=e; acc[dst,:]+=e*h[src,:]
// One wave (32 lanes) per edge: lane = d within head, loop over heads.
// ---------------------------------------------------------------------------
__global__ void edge_accum1_kernel(const long long* __restrict__ ei, int E, int N,
                                   const float* __restrict__ asrc,
                                   const float* __restrict__ adst,
                                   const float* __restrict__ m1,
                                   const float* __restrict__ h,
                                   float* __restrict__ s1,
                                   float* __restrict__ acc) {
  long long ET = (long long)E + N;
  long long e  = ((long long)blockIdx.x * blockDim.x + threadIdx.x) >> 5;
  int lane     = threadIdx.x & 31;
  if (e >= ET) return;
  int s, d;
  edge_nodes(ei, E, e, s, d);
  const float* hs = h   + (size_t)s * HID;
  float*       ad = acc + (size_t)d * HID;
  #pragma unroll
  for (int hh = 0; hh < HEADS; ++hh) {
    float l = asrc[s * HEADS + hh] + adst[d * HEADS + hh];
    l = (l > 0.f) ? l : NEG_SLOPE * l;
    float ev = __expf(l - m1[d * HEADS + hh]);
    if (lane == hh) atomicAdd(&s1[d * HEADS + hh], ev);
    atomicAdd(&ad[hh * DH + lane], ev * hs[hh * DH + lane]);
  }
}

// ---------------------------------------------------------------------------
// Node update: normalize, +b1, BatchNorm(eval), ReLU -> h2[N,128]
// ---------------------------------------------------------------------------
__global__ void node1_kernel(const float* __restrict__ acc,
                             const float* __restrict__ s1,
                             const float* __restrict__ b1,
                             const float* __restrict__ g,
                             const float* __restrict__ be,
                             const float* __restrict__ mu,
                             const float* __restrict__ var,
                             float* __restrict__ h2, int N) {
  int i = blockIdx.x * blockDim.x + threadIdx.x;
  if (i >= N * HID) return;
  int n = i >> 7, c = i & (HID - 1), hh = c >> 5;
  float v = acc[i] / s1[n * HEADS + hh] + b1[c];
  v = (v - mu[c]) * rsqrtf(var[c] + BN_EPS) * g[c] + be[c];
  h2[i] = (v > 0.f) ? v : 0.f;
}

// ---------------------------------------------------------------------------
// Layer-2 projection (tiny: 128->2) + attention coefficients, 1 thread/node
// ---------------------------------------------------------------------------
__global__ void gemm2_kernel(const float* __restrict__ h2,
                             const float* __restrict__ W2,
                             const float* __restrict__ aS,
                             const float* __restrict__ aD,
                             float* __restrict__ t,
                             float* __restrict__ as2, float* __restrict__ ad2,
                             int N) {
  int n = blockIdx.x * blockDim.x + threadIdx.x;
  if (n >= N) return;
  const float* hp = h2 + (size_t)n * HID;
  float t0 = 0.f, t1 = 0.f;
  #pragma unroll 4
  for (int c = 0; c < HID; ++c) {
    float v = hp[c];
    t0 = fmaf(v, W2[c * OUTD + 0], t0);
    t1 = fmaf(v, W2[c * OUTD + 1], t1);
  }
  t[n * OUTD + 0] = t0;
  t[n * OUTD + 1] = t1;
  as2[n] = t0 * aS[0] + t1 * aS[1];
  ad2[n] = t0 * aD[0] + t1 * aD[1];
}

__global__ void edge_max2_kernel(const long long* __restrict__ ei, int E, int N,
                                 const float* __restrict__ as2,
                                 const float* __restrict__ ad2,
                                 float* __restrict__ m2) {
  long long ET = (long long)E + N;
  long long e = (long long)blockIdx.x * blockDim.x + threadIdx.x;
  if (e >= ET) return;
  int s, d;
  edge_nodes(ei, E, e, s, d);
  float l = as2[s] + ad2[d];
  l = (l > 0.f) ? l : NEG_SLOPE * l;
  atomicMaxF(&m2[d], l);
}

__global__ void edge_accum2_kernel(const long long* __restrict__ ei, int E, int N,
                                   const float* __restrict__ as2,
                                   const float* __restrict__ ad2,
                                   const float* __restrict__ m2,
                                   const float* __restrict__ t,
                                   float* __restrict__ s2,
                                   float* __restrict__ acc2) {
  long long ET = (long long)E + N;
  long long e = (long long)blockIdx.x * blockDim.x + threadIdx.x;
  if (e >= ET) return;
  int s, d;
  edge_nodes(ei, E, e, s, d);
  float l = as2[s] + ad2[d];
  l = (l > 0.f) ? l : NEG_SLOPE * l;
  float ev = __expf(l - m2[d]);
  atomicAdd(&s2[d], ev);
  atomicAdd(&acc2[d * OUTD + 0], ev * t[s * OUTD + 0]);
  atomicAdd(&acc2[d * OUTD + 1], ev * t[s * OUTD + 1]);
}

__global__ void final_kernel(const float* __restrict__ acc2,
                             const float* __restrict__ s2,
                             const float* __restrict__ b2,
                             float* __restrict__ out, int N) {
  int i = blockIdx.x * blockDim.x + threadIdx.x;
  if (i >= N * OUTD) return;
  int n = i >> 1, j = i & 1;
  out[i] = acc2[i] / s2[n] + b2[j];
}

// ---------------------------------------------------------------------------
extern "C" void kernel_launch(void* const* d_in, const int* in_sizes, int n_in,
                              void* d_out, int out_size, void* d_ws, size_t ws_size,
                              hipStream_t stream) {
  const float*     x   = (const float*)d_in[0];
  const long long* ei  = (const long long*)d_in[1];   // int64 edge_index [2,E]
  const float*     W1  = (const float*)d_in[2];
  const float*     aS1 = (const float*)d_in[3];
  const float*     aD1 = (const float*)d_in[4];
  const float*     b1  = (const float*)d_in[5];
  const float*     bng = (const float*)d_in[6];
  const float*     bnb = (const float*)d_in[7];
  const float*     bnm = (const float*)d_in[8];
  const float*     bnv = (const float*)d_in[9];
  const float*     W2  = (const float*)d_in[10];
  const float*     aS2 = (const float*)d_in[11];
  const float*     aD2 = (const float*)d_in[12];
  const float*     b2  = (const float*)d_in[13];
  float* out = (float*)d_out;

  const int N  = in_sizes[0] / KDIM;
  const int E  = in_sizes[1] / 2;
  const long long ET = (long long)E + N;

  // ---- workspace layout (floats) ----
  float* ws = (float*)d_ws;
  float* h     = ws;  ws += (size_t)N * HID;
  float* h2    = ws;  ws += (size_t)N * HID;
  float* t     = ws;  ws += (size_t)N * OUTD;
  float* asrc1 = ws;  ws += (size_t)N * HEADS;
  float* adst1 = ws;  ws += (size_t)N * HEADS;
  float* as2   = ws;  ws += (size_t)N;
  float* ad2   = ws;  ws += (size_t)N;
  // neg-inf initialized region (contiguous): m1 then m2
  float* m1    = ws;  ws += (size_t)N * HEADS;
  float* m2    = ws;  ws += (size_t)N;
  // zero initialized region (contiguous): s1, acc1, s2, acc2
  float* s1    = ws;  ws += (size_t)N * HEADS;
  float* acc1  = ws;  ws += (size_t)N * HID;
  float* s2    = ws;  ws += (size_t)N;
  float* acc2  = ws;  ws += (size_t)N * OUTD;

  const int B = 256;
  // init softmax state
  {
    long long nneg = (long long)N * (HEADS + 1);
    long long nzer = (long long)N * (HEADS + HID + 1 + OUTD);
    fill_kernel<<<(unsigned)((nneg + B - 1) / B), B, 0, stream>>>(m1, nneg, -3.0e38f);
    fill_kernel<<<(unsigned)((nzer + B - 1) / B), B, 0, stream>>>(s1, nzer, 0.0f);
  }

  // Layer 1
  gemm1_wmma_kernel<<<(N + 15) / 16, B, 0, stream>>>(x, W1, h, N);
  alpha1_kernel<<<(N * HEADS + B - 1) / B, B, 0, stream>>>(h, aS1, aD1, asrc1, adst1, N);
  {
    long long tmax = ET * HEADS;
    edge_max1_kernel<<<(unsigned)((tmax + B - 1) / B), B, 0, stream>>>(ei, E, N, asrc1, adst1, m1);
    long long tacc = ET * 32;
    edge_accum1_kernel<<<(unsigned)((tacc + B - 1) / B), B, 0, stream>>>(
        ei, E, N, asrc1, adst1, m1, h, s1, acc1);
  }
  node1_kernel<<<(N * HID + B - 1) / B, B, 0, stream>>>(acc1, s1, b1, bng, bnb, bnm, bnv, h2, N);

  // Layer 2
  gemm2_kernel<<<(N + B - 1) / B, B, 0, stream>>>(h2, W2, aS2, aD2, t, as2, ad2, N);
  edge_max2_kernel<<<(unsigned)((ET + B - 1) / B), B, 0, stream>>>(ei, E, N, as2, ad2, m2);
  edge_accum2_kernel<<<(unsigned)((ET + B - 1) / B), B, 0, stream>>>(ei, E, N, as2, ad2, m2, t, s2, acc2);
  final_kernel<<<(N * OUTD + B - 1) / B, B, 0, stream>>>(acc2, s2, b2, out, N);
}